// MultiplexGIN_72112500899859
// MI455X (gfx1250) — compile-verified
//
#include <hip/hip_runtime.h>
#include <stdint.h>

// MultiplexGIN on MI455X (gfx1250): memory-bound gather/scatter.
// AI ~0.25 FLOP/B -> optimize the memory path, not compute:
//  - per-wave async global->LDS staging (global_load_async_to_lds_b128, ASYNCcnt)
//  - native f32 global atomic adds (resolved in the 192MB L2)
//  - fully coalesced 512B/1KB row transactions (one edge per wave32, float4/lane)
// Workspace layout (floats): deg1[N] | deg2[N] | h[N*256] | acc[N*256]  (~206 MB)

#define GIN_N 100000
#define NEG_SLOPE 0.01f

// ---------- CDNA5 async global->LDS helpers ----------
__device__ __forceinline__ void async_gather16(unsigned lds_byte_off, const float* gaddr) {
  // Per-lane: copies 16 bytes from the global address gaddr to LDS byte offset lds_byte_off, tracked by ASYNCcnt.
  asm volatile("global_load_async_to_lds_b128 %0, %1, off"
               :: "v"(lds_byte_off), "v"(gaddr)
               : "memory");
}
__device__ __forceinline__ void wait_async0() { asm volatile("s_wait_asynccnt 0x0" ::: "memory"); }
__device__ __forceinline__ void wait_async1() { asm volatile("s_wait_asynccnt 0x1" ::: "memory"); }
__device__ __forceinline__ void wait_async2() { asm volatile("s_wait_asynccnt 0x2" ::: "memory"); }

__device__ __forceinline__ void atomic_add_f32(float* p, float v) {
  unsafeAtomicAdd(p, v);   // -> global_atomic_add_u32/f32 (no-return, STOREcnt)
}

__device__ __forceinline__ float leaky(float v) {
  return v > 0.0f ? v : v * NEG_SLOPE;
}

// ---------- trivial kernels ----------
__global__ void gin_zero4(float4* __restrict__ p, int n4) {
  int i = blockIdx.x * blockDim.x + threadIdx.x;
  if (i < n4) p[i] = make_float4(0.f, 0.f, 0.f, 0.f);
}

__global__ void gin_degree(const int* __restrict__ dst, float* __restrict__ deg, int E) {
  int i = blockIdx.x * blockDim.x + threadIdx.x;
  if (i < E) atomic_add_f32(&deg[dst[i]], 1.0f);
}

// ---------- scatter: acc[dst[e]] += x[src[e]] ----------
// One edge per wave32; VECS float4 per lane (D = 128*VECS).
// Double-buffered per-wave LDS slot, async staged, 2-stage pipeline.
template <int VECS>
__global__ void gin_scatter(const float* __restrict__ x,
                            const int* __restrict__ src,
                            const int* __restrict__ dst,
                            float* __restrict__ acc, int E) {
  constexpr int D = 128 * VECS;
  __shared__ __align__(16) float buf[8 * 2 * D];   // 8 waves/block, double buffer

  const int lane = threadIdx.x & 31;
  const int wib  = threadIdx.x >> 5;               // wave-in-block [0,8)
  const int gw   = blockIdx.x * 8 + wib;           // global wave id
  const int nw   = gridDim.x * 8;

  float* slot0 = &buf[(wib * 2 + 0) * D];
  float* slot1 = &buf[(wib * 2 + 1) * D];
  // Low 32 bits of the generic address of a __shared__ object are the LDS byte offset.
  const unsigned loff0 = (unsigned)(size_t)(slot0 + lane * 4);
  const unsigned loff1 = (unsigned)(size_t)(slot1 + lane * 4);

  int e = gw;
  if (e < E) {
    int s = __builtin_amdgcn_readfirstlane(src[e]);
    const float* g = x + (size_t)s * D + lane * 4;
#pragma unroll
    for (int v = 0; v < VECS; ++v) async_gather16(loff0 + (unsigned)v * 512u, g + v * 128);
  }

  int parity = 0;
  for (; e < E; e += nw) {
    const int enext = e + nw;
    if (enext < E) {
      __builtin_prefetch(&src[enext + nw], 0, 0);  // global_prefetch_b8 on streamed indices
      __builtin_prefetch(&dst[enext + nw], 0, 0);
      int s = __builtin_amdgcn_readfirstlane(src[enext]);
      const float* g = x + (size_t)s * D + lane * 4;
      const unsigned lo = parity ? loff0 : loff1;
#pragma unroll
      for (int v = 0; v < VECS; ++v) async_gather16(lo + (unsigned)v * 512u, g + v * 128);
      // Wait until only the just-issued copies are outstanding -> current slot landed.
      if (VECS == 1) wait_async1(); else wait_async2();
    } else {
      wait_async0();
    }

    const float* cur = (parity ? slot1 : slot0) + lane * 4;
    int d = __builtin_amdgcn_readfirstlane(dst[e]);
    float* a = acc + (size_t)d * D + lane * 4;
#pragma unroll
    for (int v = 0; v < VECS; ++v) {
      const float4 val = *(const float4*)(cur + v * 128);   // ds_load_b128
      float* ap = a + v * 128;
      atomic_add_f32(ap + 0, val.x);
      atomic_add_f32(ap + 1, val.y);
      atomic_add_f32(ap + 2, val.z);
      atomic_add_f32(ap + 3, val.w);
    }
    parity ^= 1;
  }
}

// ---------- finalize: out[:, off:off+D] = leaky(x + acc/max(deg,1)) ----------
template <int D>
__global__ void gin_finalize(const float* __restrict__ xin,   // stride D
                             const float* __restrict__ acc,   // stride D
                             const float* __restrict__ deg,
                             float* __restrict__ out, int out_stride, int out_off) {
  const int idx = blockIdx.x * blockDim.x + threadIdx.x;
  constexpr int Q = D / 4;
  if (idx >= GIN_N * Q) return;
  const int n  = idx / Q;
  const int f4 = idx - n * Q;
  const float inv = 1.0f / fmaxf(deg[n], 1.0f);
  const float4 xv = *(const float4*)(xin + (size_t)n * D + f4 * 4);
  const float4 av = *(const float4*)(acc + (size_t)n * D + f4 * 4);
  float4 y;
  y.x = leaky(fmaf(av.x, inv, xv.x));
  y.y = leaky(fmaf(av.y, inv, xv.y));
  y.z = leaky(fmaf(av.z, inv, xv.z));
  y.w = leaky(fmaf(av.w, inv, xv.w));
  *(float4*)(out + (size_t)n * out_stride + out_off + f4 * 4) = y;
}

// ---------- host ----------
static inline unsigned cdiv(long long n, int b) { return (unsigned)((n + b - 1) / b); }

extern "C" void kernel_launch(void* const* d_in, const int* in_sizes, int n_in,
                              void* d_out, int out_size, void* d_ws, size_t ws_size,
                              hipStream_t stream) {
  const float* x  = (const float*)d_in[0];
  const int* e1s  = (const int*)d_in[1];
  const int* e1d  = (const int*)d_in[2];
  const int* e2s  = (const int*)d_in[3];
  const int* e2d  = (const int*)d_in[4];
  const int  E1   = in_sizes[1];
  const int  E2   = in_sizes[3];
  float* out = (float*)d_out;

  const long long N = GIN_N;
  float* deg1 = (float*)d_ws;
  float* deg2 = deg1 + N;
  float* h    = deg2 + N;                 // [N, 256]
  float* acc  = h + N * 256;              // [N, 256] reused across 4 phases

  const dim3 B(256);
  const dim3 SCAT(2048);                  // 2048 blocks * 8 waves = 16384 waves, grid-strided

  // degrees (shared by both layers)
  gin_zero4<<<cdiv(2 * N / 4, 256), B, 0, stream>>>((float4*)deg1, (int)(2 * N / 4));
  gin_degree<<<cdiv(E1, 256), B, 0, stream>>>(e1d, deg1, E1);
  gin_degree<<<cdiv(E2, 256), B, 0, stream>>>(e2d, deg2, E2);

  // ---- layer 1, edges1 -> h[:, 0:128]
  gin_zero4<<<cdiv(N * 128 / 4, 256), B, 0, stream>>>((float4*)acc, (int)(N * 128 / 4));
  gin_scatter<1><<<SCAT, B, 0, stream>>>(x, e1s, e1d, acc, E1);
  gin_finalize<128><<<cdiv(N * 32, 256), B, 0, stream>>>(x, acc, deg1, h, 256, 0);

  // ---- layer 1, edges2 -> h[:, 128:256]
  gin_zero4<<<cdiv(N * 128 / 4, 256), B, 0, stream>>>((float4*)acc, (int)(N * 128 / 4));
  gin_scatter<1><<<SCAT, B, 0, stream>>>(x, e2s, e2d, acc, E2);
  gin_finalize<128><<<cdiv(N * 32, 256), B, 0, stream>>>(x, acc, deg2, h, 256, 128);

  // ---- layer 2, edges1 -> out[:, 0:256]
  gin_zero4<<<cdiv(N * 256 / 4, 256), B, 0, stream>>>((float4*)acc, (int)(N * 256 / 4));
  gin_scatter<2><<<SCAT, B, 0, stream>>>(h, e1s, e1d, acc, E1);
  gin_finalize<256><<<cdiv(N * 64, 256), B, 0, stream>>>(h, acc, deg1, out, 512, 0);

  // ---- layer 2, edges2 -> out[:, 256:512]
  gin_zero4<<<cdiv(N * 256 / 4, 256), B, 0, stream>>>((float4*)acc, (int)(N * 256 / 4));
  gin_scatter<2><<<SCAT, B, 0, stream>>>(h, e2s, e2d, acc, E2);
  gin_finalize<256><<<cdiv(N * 64, 256), B, 0, stream>>>(h, acc, deg2, out, 512, 256);
}